// MedianFilterAugment_15058155340053
// MI455X (gfx1250) — compile-verified
//
#include <hip/hip_runtime.h>
#include <hip/hip_bf16.h>

// Per-sample variable-size 1D median filter (window <= 19, reflect pad).
//
// The per-sample window size s is uniform across each (b, c, :) row, hence
// uniform per block. Instead of the reference's sentinel-padded sort of a
// fixed 19-window (~103 comparators always), we dispatch via a scalar
// switch(s) to a Batcher odd-even mergesort of exactly s elements and return
// the lower median, sorted index (s-1)/2. Average comparator count over
// s in [1,18] is ~35 -> ~3x less VALU work (this kernel is VALU-bound:
// HBM floor is ~2.9us at 23.3 TB/s, the 19-sort costs ~10x that in VALU).
//
// CDNA5 path: tile+halo staged to LDS with async global->LDS copies
// (ASYNCcnt + s_wait_asynccnt); interior tiles use 16B-aligned
// global_load_async_to_lds_b128, row-edge tiles the b32 reflect fallback.

#define K_MAX  19
#define MHALF  9
#define BLOCK  256
#define PAD    12                      // halo padded 9->12 for 16B alignment
#define NLOAD  (BLOCK + 2 * PAD)       // 280 floats = 70 x 16B

__device__ __forceinline__ void cswap(float& a, float& b) {
    float lo = fminf(a, b);
    float hi = fmaxf(a, b);
    a = lo;
    b = hi;
}

// Batcher odd-even mergesort network for arbitrary N, fully unrolled.
template <int N>
__device__ __forceinline__ void batcher_sort(float (&v)[N]) {
    #pragma unroll
    for (int p = 1; p < N; p <<= 1) {
        #pragma unroll
        for (int k = p; k >= 1; k >>= 1) {
            #pragma unroll
            for (int j = k & (p - 1); j + k < N; j += 2 * k) {
                #pragma unroll
                for (int i = 0; i < k; ++i) {
                    if (i + j + k < N) {
                        if ((i + j) / (2 * p) == (i + j + k) / (2 * p)) {
                            cswap(v[i + j], v[i + j + k]);
                        }
                    }
                }
            }
        }
    }
}

// Lower median of S consecutive LDS values starting at w0.
template <int S>
__device__ __forceinline__ float medianS(const float* __restrict__ w0) {
    float v[S];
    #pragma unroll
    for (int j = 0; j < S; ++j) v[j] = w0[j];
    batcher_sort<S>(v);
    return v[(S - 1) / 2];
}

__global__ __launch_bounds__(BLOCK) void median19_kernel(
    const float* __restrict__ x,
    const int*   __restrict__ sizes,
    float*       __restrict__ out,
    int T)
{
    __shared__ __align__(16) float tile[NLOAD];

    const int tid   = threadIdx.x;
    const int row   = blockIdx.y;        // row = b*C + c, C == 2
    const int b     = row >> 1;
    const int tile0 = blockIdx.x * BLOCK;
    const long long rowbase = (long long)row * (long long)T;
    const float* rowp = x + rowbase;

    const unsigned lds_base = (unsigned)(uintptr_t)&tile[0];

    // ---- Stage tile + halo into LDS via gfx1250 async global->LDS copies ----
    if (tile0 - PAD >= 0 && tile0 + BLOCK + PAD <= T) {
        // Interior tile: contiguous, 16B-aligned. 70 b128 async copies.
        if (tid < NLOAD / 4) {
            unsigned long long gaddr =
                (unsigned long long)(uintptr_t)(rowp + (tile0 - PAD) + 4 * tid);
            unsigned laddr = lds_base + (unsigned)tid * 16u;
            asm volatile("global_load_async_to_lds_b128 %0, %1, off"
                         :: "v"(laddr), "v"(gaddr)
                         : "memory");
        }
    } else {
        // Row-edge tile: reflect indexing breaks contiguity -> b32 copies.
        #pragma unroll
        for (int rep = 0; rep < 2; ++rep) {
            int pos = tid + rep * BLOCK;
            if (pos < NLOAD) {
                int g = tile0 - PAD + pos;
                g = (g < 0) ? -g : g;                       // reflect left
                g = (g > T - 1) ? (2 * T - 2 - g) : g;      // reflect right
                unsigned long long gaddr =
                    (unsigned long long)(uintptr_t)(rowp + g);
                unsigned laddr = lds_base + (unsigned)pos * 4u;
                asm volatile("global_load_async_to_lds_b32 %0, %1, off"
                             :: "v"(laddr), "v"(gaddr)
                             : "memory");
            }
        }
    }
    asm volatile("s_wait_asynccnt 0" ::: "memory");
    __syncthreads();

    // ---- Per-sample window size (uniform per block -> scalar branch) ----
    int s = sizes[b];
    s = (s < 1) ? 1 : ((s > K_MAX) ? K_MAX : s);

    // LDS position p holds global element (tile0 - PAD + p). The window of
    // size S for output t = tile0 + tid starts (S-1)/2 elements left of
    // center: LDS index tid + PAD - (S-1)/2.
    const float* wbase = tile + tid + PAD;
    float r;
    #define MCASE(S) case S: r = medianS<S>(wbase - ((S - 1) / 2)); break;
    switch (s) {
        MCASE(1)  MCASE(2)  MCASE(3)  MCASE(4)  MCASE(5)
        MCASE(6)  MCASE(7)  MCASE(8)  MCASE(9)  MCASE(10)
        MCASE(11) MCASE(12) MCASE(13) MCASE(14) MCASE(15)
        MCASE(16) MCASE(17) MCASE(18)
        default:  r = medianS<19>(wbase - 9); break;
    }
    #undef MCASE

    const int t = tile0 + tid;
    if (t < T) out[rowbase + t] = r;
}

extern "C" void kernel_launch(void* const* d_in, const int* in_sizes, int n_in,
                              void* d_out, int out_size, void* d_ws, size_t ws_size,
                              hipStream_t stream) {
    const float* x     = (const float*)d_in[0];
    const int*   sizes = (const int*)d_in[1];
    float*       out   = (float*)d_out;

    const int B  = in_sizes[1];          // 16
    const int BC = B * 2;                // C == 2
    const int T  = in_sizes[0] / BC;     // 262144

    dim3 block(BLOCK);
    dim3 grid((T + BLOCK - 1) / BLOCK, BC);
    hipLaunchKernelGGL(median19_kernel, grid, block, 0, stream,
                       x, sizes, out, T);
}